// ExpectSpatialSoftmax_13692355739806
// MI455X (gfx1250) — compile-verified
//
#include <hip/hip_runtime.h>
#include <math.h>

typedef __attribute__((ext_vector_type(4))) unsigned int u32x4;
typedef __attribute__((ext_vector_type(8))) unsigned int u32x8;

#define HW_N      16384                 // H*W per (b,c) row
#define CHUNK     2048                  // floats per TDM chunk (8 KB)
#define NCHUNK    (HW_N / CHUNK)        // 8 chunks per row
#define NTHREADS  256                   // 8 wave32s
#define PER_THR   (CHUNK / NTHREADS)    // 8 floats / thread / chunk
#define NBUF      4                     // LDS ring buffers (32 KB)
#define PF        3                     // TDM prefetch distance (3 in flight)

// Issue one TDM descriptor: copies CHUNK contiguous f32 from global memory to the LDS byte offset in lds_off.
// D# layout per CDNA5 ISA ch.8 (groups 0 and 1; groups 2/3 NULL => <=2D tensor).
__device__ __forceinline__ void tdm_load_chunk(const float* gsrc, unsigned lds_off) {
  unsigned long long ga = (unsigned long long)(uintptr_t)gsrc;
  u32x4 g0;
  g0[0] = 1u;                                                  // count=1, user mode
  g0[1] = lds_off;                                             // lds_addr [63:32]
  g0[2] = (unsigned)(ga & 0xFFFFFFFFull);                      // global_addr lo
  g0[3] = (unsigned)((ga >> 32) & 0x01FFFFFFu) | (2u << 30);   // addr[56:32] | type=2
  u32x8 g1;
  g1[0] = 2u << 16;                                            // data_size = 4 bytes
  g1[1] = ((unsigned)CHUNK & 0xFFFFu) << 16;                   // tensor_dim0 lo16 @[63:48]
  g1[2] = (((unsigned)CHUNK >> 16) & 0xFFFFu) | (1u << 16);    // dim0 hi16 | tensor_dim1=1
  g1[3] = ((unsigned)CHUNK) << 16;                             // tile_dim0 @[127:112]
  g1[4] = 1u;                                                  // tile_dim1 = 1
  g1[5] = (unsigned)CHUNK;                                     // tensor_dim0_stride lo32
  g1[6] = 0u;
  g1[7] = 0u;
  asm volatile("tensor_load_to_lds %0, %1" :: "s"(g0), "s"(g1) : "memory");
}

// One software-pipelined chunk. C is compile-time so the tensorcnt wait target
// is an immediate: with PF DMAs in flight and in-order TDM completion,
// TENSORcnt <= min(PF-1, NCHUNK-1-C) guarantees chunk C has landed.
template <int C>
__device__ __forceinline__ void process_chunk(const float* __restrict__ rowp,
                                              float (&buf)[NBUF][CHUNK],
                                              unsigned lds0, int t, bool w0,
                                              float& m, float& s, float& sx, float& sy) {
  constexpr int rem   = NCHUNK - 1 - C;
  constexpr int waitn = rem < (PF - 1) ? rem : (PF - 1);
  if (w0) __builtin_amdgcn_s_wait_tensorcnt(waitn);
  __syncthreads();   // chunk C visible to all waves; also orders prior reads of
                     // buf[(C+PF)%NBUF] (iteration C-1) before the issue below
  if constexpr (C + PF < NCHUNK) {
    if (w0)
      tdm_load_chunk(rowp + (size_t)(C + PF) * CHUNK,
                     lds0 + (unsigned)(((C + PF) % NBUF) * CHUNK * sizeof(float)));
  }

  // 8 contiguous floats per thread -> 2x ds_load_b128
  const float4* bp = reinterpret_cast<const float4*>(&buf[C % NBUF][t * PER_THR]);
  float4 a0 = bp[0], a1 = bp[1];
  float v[PER_THR] = {a0.x, a0.y, a0.z, a0.w, a1.x, a1.y, a1.z, a1.w};

  float cm = fmaxf(fmaxf(fmaxf(v[0], v[1]), fmaxf(v[2], v[3])),
                   fmaxf(fmaxf(v[4], v[5]), fmaxf(v[6], v[7])));
  float nm  = fmaxf(m, cm);
  float scl = __expf(m - nm);                   // exp(-inf)=0 on first chunk
  s *= scl; sx *= scl; sy *= scl;

  const float csc = 2.0f / 127.0f;              // linspace(-1,1,128) step
  const int base = C * CHUNK + t * PER_THR;
  #pragma unroll
  for (int k = 0; k < PER_THR; ++k) {
    int n = base + k;
    float cx = fmaf((float)(n & 127), csc, -1.0f);  // x varies along width j
    float cy = fmaf((float)(n >> 7), csc, -1.0f);   // y varies along height i
    float p  = __expf(v[k] - nm);
    s  += p;
    sx  = fmaf(p, cx, sx);
    sy  = fmaf(p, cy, sy);
  }
  m = nm;
}

__global__ __launch_bounds__(NTHREADS)
void spatial_softmax_kp_kernel(const float* __restrict__ x, float* __restrict__ out) {
  __shared__ __align__(16) float buf[NBUF][CHUNK];   // 32 KB ring buffer
  __shared__ float red[8][4];                        // per-wave partials

  const int row = blockIdx.x;                        // b*C + c
  const int t   = threadIdx.x;
  const bool w0 = (t >> 5) == 0;                     // wave 0 drives the TDM
  const float* rowp = x + (size_t)row * HW_N;

  // LDS byte offset of the ring (low 32 bits of generic ptr == LDS offset)
  const unsigned lds0 = (unsigned)(uintptr_t)&buf[0][0];

  if (w0) {                                          // prime PF chunks
    tdm_load_chunk(rowp,             lds0);
    tdm_load_chunk(rowp + CHUNK,     lds0 + 1u * CHUNK * sizeof(float));
    tdm_load_chunk(rowp + 2 * CHUNK, lds0 + 2u * CHUNK * sizeof(float));
  }

  float m = -INFINITY, s = 0.f, sx = 0.f, sy = 0.f;
  process_chunk<0>(rowp, buf, lds0, t, w0, m, s, sx, sy);
  process_chunk<1>(rowp, buf, lds0, t, w0, m, s, sx, sy);
  process_chunk<2>(rowp, buf, lds0, t, w0, m, s, sx, sy);
  process_chunk<3>(rowp, buf, lds0, t, w0, m, s, sx, sy);
  process_chunk<4>(rowp, buf, lds0, t, w0, m, s, sx, sy);
  process_chunk<5>(rowp, buf, lds0, t, w0, m, s, sx, sy);
  process_chunk<6>(rowp, buf, lds0, t, w0, m, s, sx, sy);
  process_chunk<7>(rowp, buf, lds0, t, w0, m, s, sx, sy);

  // wave32 merge-reduction of (m,s,sx,sy)
  #pragma unroll
  for (int off = 16; off >= 1; off >>= 1) {
    float om  = __shfl_xor(m,  off, 32);
    float os  = __shfl_xor(s,  off, 32);
    float osx = __shfl_xor(sx, off, 32);
    float osy = __shfl_xor(sy, off, 32);
    float nm = fmaxf(m, om);
    float a  = __expf(m  - nm);
    float b  = __expf(om - nm);
    s  = s  * a + os  * b;
    sx = sx * a + osx * b;
    sy = sy * a + osy * b;
    m  = nm;
  }
  if ((t & 31) == 0) {
    int w = t >> 5;
    red[w][0] = m; red[w][1] = s; red[w][2] = sx; red[w][3] = sy;
  }
  __syncthreads();

  if (t < 8) {
    float m2 = red[t][0], s2 = red[t][1], sx2 = red[t][2], sy2 = red[t][3];
    #pragma unroll
    for (int off = 4; off >= 1; off >>= 1) {      // partners stay within lanes 0..7
      float om  = __shfl_xor(m2,  off, 32);
      float os  = __shfl_xor(s2,  off, 32);
      float osx = __shfl_xor(sx2, off, 32);
      float osy = __shfl_xor(sy2, off, 32);
      float nm = fmaxf(m2, om);
      float a  = __expf(m2 - nm);
      float b  = __expf(om - nm);
      s2  = s2  * a + os  * b;
      sx2 = sx2 * a + osx * b;
      sy2 = sy2 * a + osy * b;
      m2  = nm;
    }
    if (t == 0) {
      float inv = 1.0f / s2;
      out[row * 2 + 0] = sx2 * inv;
      out[row * 2 + 1] = sy2 * inv;
    }
  }
}

extern "C" void kernel_launch(void* const* d_in, const int* in_sizes, int n_in,
                              void* d_out, int out_size, void* d_ws, size_t ws_size,
                              hipStream_t stream) {
  const float* x = (const float*)d_in[0];
  float* out = (float*)d_out;
  int rows = in_sizes[0] / HW_N;                  // 32*128 = 4096
  hipLaunchKernelGGL(spatial_softmax_kp_kernel, dim3(rows), dim3(NTHREADS), 0, stream,
                     x, out);
}